// SLP_Pooling_Module_27041114096403
// MI455X (gfx1250) — compile-verified
//
#include <hip/hip_runtime.h>
#include <hip/hip_bf16.h>

#define B_   16
#define N_   4096
#define S_   1024
#define K_   32
#define DIN  64
#define DINC 67
#define DOUT 64
#define MAXV 1e10f

typedef __attribute__((ext_vector_type(2))) float v2f;
typedef __attribute__((ext_vector_type(8))) float v8f;

// ---------------------------------------------------------------------------
// Kernel 1: comb = x_complete@w1^T + bias (+ x@w2^T if use_x), xp = x@w2^T
// One 16-row tile (of B*N rows) per wave, C=64 as 4 WMMA column tiles.
// fp32 WMMA (V_WMMA_F32_16X16X4_F32) keeps exact fp32 math vs reference.
// LDS weights stored as k-pairs (float2) so each B fragment is ONE ds_load_b64
// directly into the WMMA operand pair (no reassembly movs).
// ---------------------------------------------------------------------------
__global__ __launch_bounds__(256) void slp_linear_wmma(
    const float* __restrict__ x, const float* __restrict__ xcmp,
    const float* __restrict__ w1, const float* __restrict__ w2,
    const float* __restrict__ bias, const int* __restrict__ useXp,
    float* __restrict__ comb, float* __restrict__ xp_out)
{
    // lwXp[kpair][c] = { w[2*kpair][c], w[2*kpair+1][c] }
    // w1 padded to K=68 (34 pairs): pair 33 = { w[66][c], 0 }.
    __shared__ v2f lw1p[34 * 64];   // 17408 B
    __shared__ v2f lw2p[32 * 64];   // 16384 B
    __shared__ float lb[64];

    const int tid = threadIdx.x;
    for (int i = tid; i < 34 * 64; i += 256) {
        const int kp = i >> 6, c = i & 63, k = kp * 2;
        v2f v;
        v.x = (k < DINC)     ? w1[c * DINC + k]     : 0.0f;
        v.y = (k + 1 < DINC) ? w1[c * DINC + k + 1] : 0.0f;
        lw1p[i] = v;
    }
    for (int i = tid; i < 32 * 64; i += 256) {
        const int kp = i >> 6, c = i & 63, k = kp * 2;
        v2f v = *(const v2f*)(w2 + c * DIN + k);    // k even -> 8B aligned
        lw2p[i] = v;
    }
    if (tid < 64) lb[tid] = bias[tid];
    __syncthreads();

    const int lane  = tid & 31;
    const int wave  = tid >> 5;
    const int tile  = blockIdx.x * 8 + wave;    // 16-row tile over B*N rows
    const int r0    = tile * 16;
    const int m     = lane & 15;                // M (row within tile) for A-frag
    const int khalf = (lane < 16) ? 0 : 2;      // K sub-pair per half-wave (A)
    const int kpoff = (lane < 16) ? 0 : 1;      // k-pair index offset (B)
    const long rowA = (long)(r0 + m);

    v8f accP[4];   // xp  = x @ w2^T
    v8f accQ[4];   // xc  = x_complete @ w1^T
#pragma unroll
    for (int ct = 0; ct < 4; ++ct) { accP[ct] = {}; accQ[ct] = {}; }

    // GEMM2: K = 64 -> 16 steps of 4
#pragma unroll
    for (int kk = 0; kk < 16; ++kk) {
        const int k0 = kk * 4 + khalf;
        const int kp = kk * 2 + kpoff;
        v2f a = *(const v2f*)(x + rowA * DIN + k0);   // row stride 64 -> aligned
#pragma unroll
        for (int ct = 0; ct < 4; ++ct) {
            v2f b = lw2p[kp * 64 + ct * 16 + m];      // single ds_load_b64
            accP[ct] = __builtin_amdgcn_wmma_f32_16x16x4_f32(
                false, a, false, b, (short)0, accP[ct], false, false);
        }
    }

    // GEMM1: K = 67 -> 17 steps of 4 (tail zero-padded via LDS pair table)
#pragma unroll
    for (int kk = 0; kk < 17; ++kk) {
        const int k0 = kk * 4 + khalf;
        const int kp = kk * 2 + kpoff;
        v2f a;
        a.x = (k0 < DINC)     ? xcmp[rowA * DINC + k0]     : 0.0f;
        a.y = (k0 + 1 < DINC) ? xcmp[rowA * DINC + k0 + 1] : 0.0f;
#pragma unroll
        for (int ct = 0; ct < 4; ++ct) {
            v2f b = lw1p[kp * 64 + ct * 16 + m];
            accQ[ct] = __builtin_amdgcn_wmma_f32_16x16x4_f32(
                false, a, false, b, (short)0, accQ[ct], false, false);
        }
    }

    // C/D layout: VGPR j -> row (j + 0/8 by half-wave), col = lane&15 in c-tile
    const int useX  = *useXp;
    const int rbase = r0 + ((lane < 16) ? 0 : 8);
#pragma unroll
    for (int ct = 0; ct < 4; ++ct) {
        const int c  = ct * 16 + m;
        const float bc = lb[c];
#pragma unroll
        for (int j = 0; j < 8; ++j) {
            const long row = (long)(rbase + j);
            const float p  = accP[ct][j];
            float q        = accQ[ct][j] + bc;
            if (useX) q += p;
            comb[row * DOUT + c]   = q;
            xp_out[row * DOUT + c] = p;
        }
    }
}

// ---------------------------------------------------------------------------
// Kernel 2: per (b,s) query, gather K=32 neighbor rows of comb (L2-resident,
// comb is 16.8 MB << 192 MB L2), subtract center xp row, masked max over K.
// One wave per query, 2 channels/lane. Indices are wave-uniform -> branches
// are non-divergent.
// ---------------------------------------------------------------------------
__global__ __launch_bounds__(256) void slp_gather_max(
    const long long* __restrict__ idx, const float* __restrict__ comb,
    const float* __restrict__ xp, const int* __restrict__ useXp,
    float* __restrict__ out)
{
    const int lane = threadIdx.x & 31;
    const int wv   = blockIdx.x * 8 + (threadIdx.x >> 5);  // flat (b,s)
    const int b    = wv / S_;
    const int s    = wv % S_;
    const long long* ip = idx + ((long)b * S_ + s) * K_;
    const int useX = *useXp;
    const long base = (long)b * N_ * DOUT;
    const int  c2   = lane * 2;

    float cx = 0.0f, cy = 0.0f;
    if (useX) {
        const long i0 = (long)ip[0];                 // guaranteed valid (>=0)
        const v2f cr = *(const v2f*)(xp + base + i0 * DOUT + c2);
        cx = cr.x; cy = cr.y;
    }

    float mx = -3.4e38f, my = -3.4e38f;
    for (int k = 0; k < K_; ++k) {
        const long id = (long)ip[k];
        float vx, vy;
        if (id >= 0) {
            const v2f r = *(const v2f*)(comb + base + id * DOUT + c2);
            vx = r.x - cx;
            vy = r.y - cy;
        } else {
            // gathered pad row is zero: (0 - center) - MAX_VALUE
            vx = -cx - MAXV;
            vy = -cy - MAXV;
        }
        mx = fmaxf(mx, vx);
        my = fmaxf(my, vy);
    }

    v2f mo; mo.x = mx; mo.y = my;
    *(v2f*)(out + ((long)b * S_ + s) * DOUT + c2) = mo;
}

// ---------------------------------------------------------------------------
extern "C" void kernel_launch(void* const* d_in, const int* in_sizes, int n_in,
                              void* d_out, int out_size, void* d_ws, size_t ws_size,
                              hipStream_t stream) {
    const float*     x    = (const float*)d_in[0];
    const float*     xcmp = (const float*)d_in[1];
    const long long* idx  = (const long long*)d_in[2];
    const float*     w1   = (const float*)d_in[3];
    const float*     w2   = (const float*)d_in[4];
    const float*     bias = (const float*)d_in[5];
    const int*       useX = (const int*)d_in[6];

    float* comb = (float*)d_ws;                          // (B*N, 64)
    float* xp   = comb + (size_t)B_ * N_ * DOUT;         // (B*N, 64)
    float* out  = (float*)d_out;                         // (B, S, 64)

    // 8 waves/block, 16 rows/wave -> 128 rows/block; 65536 rows -> 512 blocks
    slp_linear_wmma<<<(B_ * N_) / 128, 256, 0, stream>>>(
        x, xcmp, w1, w2, bias, useX, comb, xp);

    // 8 queries/block; B*S = 16384 -> 2048 blocks
    slp_gather_max<<<(B_ * S_) / 8, 256, 0, stream>>>(
        idx, comb, xp, useX, out);
}